// GraphConv_76441827934787
// MI455X (gfx1250) — compile-verified
//

#include <hip/hip_runtime.h>
#include <hip/hip_bf16.h>

// MI455X / gfx1250 (wave32, WMMA). Gather-bound graph conv:
//  - kernel 1: local 3x3 reflect conv, writes 0.5*conv + bias into d_out
//  - kernel 2: non-local branch; per-wave tiles of 16 rows run the per-row
//    3x3 matvec chains via V_WMMA_F32_16X16X4_F32 (full fp32). Weight
//    operands are built branchlessly from an LDS table; each wave processes
//    4 tiles to amortize setup and keep 4 gather streams in flight.
//    All indices are 32-bit so gathers use saddr+voffset addressing.

typedef float v2f __attribute__((ext_vector_type(2)));
typedef float v8f __attribute__((ext_vector_type(8)));

__device__ __forceinline__ float bperm_f(int src_lane, float v) {
    return __int_as_float(__builtin_amdgcn_ds_bpermute(src_lane << 2, __float_as_int(v)));
}

__device__ __forceinline__ float leaky(float x) { return x > 0.0f ? x : 0.01f * x; }

// ---------------------------------------------------------------------------
// Local branch: 3x3 conv with reflect padding; out = 0.5*conv + bias.
// blockIdx = (b*3+o)*256 + row ; threadIdx.x = col (o,b,row wave-uniform).
// ---------------------------------------------------------------------------
__global__ __launch_bounds__(256) void conv_local_kernel(
    const float* __restrict__ h, const float* __restrict__ cw,
    const float* __restrict__ bias, float* __restrict__ out)
{
    const unsigned hh = blockIdx.x & 255u;
    const unsigned bo = blockIdx.x >> 8;     // b*3 + o, 0..23
    const unsigned o  = bo % 3u;
    const unsigned w  = threadIdx.x;

    float acc = 0.0f;
#pragma unroll
    for (unsigned c = 0; c < 3; ++c) {
        const float* hp = h + (((bo / 3u) * 3u + c) << 16);
        const float* wp = cw + (o * 3u + c) * 9u;
#pragma unroll
        for (int dy = -1; dy <= 1; ++dy) {
            int y = (int)hh + dy;
            y = (y < 0) ? -y : ((y > 255) ? 510 - y : y);
#pragma unroll
            for (int dx = -1; dx <= 1; ++dx) {
                int x = (int)w + dx;
                x = (x < 0) ? -x : ((x > 255) ? 510 - x : x);
                acc += hp[((unsigned)y << 8) + (unsigned)x] * wp[(dy + 1) * 3 + (dx + 1)];
            }
        }
    }
    out[(blockIdx.x << 8) + w] = 0.5f * acc + bias[o];
}

// ---------------------------------------------------------------------------
// Non-local branch. One wave handles TILES tiles of 16 rows each; a tile's
// flat-f range of 48 covers exactly 6 exclusively-owned output cells.
// Constants: N = 1<<16, N*K = 1<<19, C = 3, K = 8. Max flat f = 12,582,912.
// ---------------------------------------------------------------------------
#define TILES 4

// LDS weight-table offsets (+ zero slots at the tail)
#define OW0 0
#define OB0 9
#define OWL 12
#define OBL 39
#define OWR 48
#define OBR 75
#define OWK 84
#define OBK 93
#define OZ  96   // slots 96..103 are 0.0f

__global__ __launch_bounds__(256) void graph_branch_kernel(
    const float* __restrict__ h, const int* __restrict__ edge,
    const float* __restrict__ W0p, const float* __restrict__ b0p,
    const float* __restrict__ WLp, const float* __restrict__ bLp,
    const float* __restrict__ WRp, const float* __restrict__ bRp,
    const float* __restrict__ Wkp, const float* __restrict__ bkp,
    float* __restrict__ out)
{
    __shared__ float wtab[104];
    __shared__ float lds_go[8][48];

    const unsigned tid  = threadIdx.x;
    const unsigned lane = tid & 31u;
    const unsigned wave = tid >> 5;
    const unsigned s    = lane & 15u;    // sample within tile
    const bool lo       = lane < 16u;

    // ---- stage all weights/biases into LDS once per block ----
    if (tid < 104u) {
        float v = 0.0f;
        if      (tid < 9u)  v = W0p[tid];
        else if (tid < 12u) v = b0p[tid - 9u];
        else if (tid < 39u) v = WLp[tid - 12u];
        else if (tid < 48u) v = bLp[tid - 39u];
        else if (tid < 75u) v = WRp[tid - 48u];
        else if (tid < 84u) v = bRp[tid - 75u];
        else if (tid < 93u) v = Wkp[tid - 84u];
        else if (tid < 96u) v = bkp[tid - 93u];
        wtab[tid] = v;                   // 96..103 stay 0.0f (zero slots)
    }
    __syncthreads();

    // ---- wave-invariant WMMA operands, built branchlessly from LDS ----
    const int m  = (int)s;               // A-matrix row owned by this lane
    const int kb = lo ? 0 : 2;           // K base for this half-wave

    v2f a1;                              // W0ext (16x4)
    a1.x = wtab[(m < 3) ? (OW0 + m * 3 + kb) : OZ];
    a1.y = wtab[(lo && m < 3) ? (OW0 + m * 3 + 1) : OZ];
    v8f c1;                              // b0 broadcast (rows 0..2)
#pragma unroll
    for (int j = 0; j < 8; ++j) c1[j] = wtab[(lo && j < 3) ? (OB0 + j) : OZ];

    v2f a2a;                             // rows 0..8 = WL, rows 9..11 = Wk
    const int ia = (m < 9) ? (OWL + m * 3) : ((m < 12) ? (OWK + (m - 9) * 3) : -1);
    a2a.x = wtab[(ia >= 0) ? (ia + kb) : OZ];
    a2a.y = wtab[(lo && ia >= 0) ? (ia + 1) : OZ];
    v8f c2a;                             // rows 0..8 = bL, rows 9..11 = bk
#pragma unroll
    for (int j = 0; j < 8; ++j) {
        int mm = lo ? j : j + 8;
        c2a[j] = wtab[(mm < 9) ? (OBL + mm) : ((mm < 12) ? (OBK + mm - 9) : OZ)];
    }

    v2f a2b;                             // rows 0..8 = WR
    a2b.x = wtab[(m < 9) ? (OWR + m * 3 + kb) : OZ];
    a2b.y = wtab[(lo && m < 9) ? (OWR + m * 3 + 1) : OZ];
    v8f c2b;                             // rows 0..8 = bR
#pragma unroll
    for (int j = 0; j < 8; ++j) {
        int mm = lo ? j : j + 8;
        c2b[j] = wtab[(mm < 9) ? (OBR + mm) : OZ];
    }

    const unsigned tileBase = (blockIdx.x * 8u + wave) * TILES;

#pragma unroll
    for (unsigned it = 0; it < TILES; ++it) {
        const unsigned tile = tileBase + it;
        const unsigned row  = tile * 16u + s;

        // ---- gather lt/vt: flat reinterpretation of (B,C,N,K) ----
        // lane<16 fetches features 0,1 of its sample; lane>=16 feature 2.
        const unsigned f0 = 3u * row + (lo ? 0u : 2u);
        const unsigned f1 = 3u * row + 1u;
        const unsigned q0 = f0 >> 19, rm0 = f0 & 0x7FFFFu;
        const unsigned q1 = f1 >> 19, rm1 = f1 & 0x7FFFFu;
        const unsigned e0 = (q0 / 3u << 19) + rm0;     // edge element index
        const unsigned e1 = (q1 / 3u << 19) + rm1;
        const unsigned hb0 = q0 << 16, hb1 = q1 << 16; // (b,c)-plane base in h
        __builtin_prefetch(edge + e0 + 48u, 0, 0);     // next tile's edge stream
        const unsigned i0 = (unsigned)edge[e0];
        const unsigned i1 = (unsigned)edge[e1];
        const float vtA = h[hb0 + i0];
        const float ltA = vtA - h[hb0 + (rm0 >> 3)];
        float vtB = h[hb1 + i1];
        float ltB = vtB - h[hb1 + (rm1 >> 3)];
        vtB = lo ? vtB : 0.0f;           // K=3 slot of B operand must be 0
        ltB = lo ? ltB : 0.0f;

        // ---- GEMM1: thetaT = leakyrelu(W0ext @ ltT + b0) ----
        v2f b1; b1.x = ltA; b1.y = ltB;
        v8f d1 = __builtin_amdgcn_wmma_f32_16x16x4_f32(false, a1, false, b1,
                                                       (short)0, c1, false, false);
        const float th0 = leaky(d1[0]);
        const float th1 = leaky(d1[1]);
        const float th2 = leaky(d1[2]);

        // thetaT into B layout: unconditional bpermute, then cndmask select
        const float th2bp = bperm_f((int)(lane & 15u), th2);
        v2f b2;
        b2.x = lo ? th0 : th2bp;
        b2.y = lo ? th1 : 0.0f;

        // ---- GEMM2a: [WL;Wk] @ thetaT + [bL;bk] ----
        v8f d2a = __builtin_amdgcn_wmma_f32_16x16x4_f32(false, a2a, false, b2,
                                                        (short)0, c2a, false, false);
        // ---- GEMM2b: WR @ thetaT + bR ----
        v8f d2b = __builtin_amdgcn_wmma_f32_16x16x4_f32(false, a2b, false, b2,
                                                        (short)0, c2b, false, false);

        // ---- pull rows m>=8 (upper half-wave of D) down to sample lanes ----
        const float tL8  = bperm_f((int)lane + 16, d2a[0]);  // thetaL row 8
        const float kap0 = bperm_f((int)lane + 16, d2a[1]);  // kappa rows 9..11
        const float kap1 = bperm_f((int)lane + 16, d2a[2]);
        const float kap2 = bperm_f((int)lane + 16, d2a[3]);
        const float tR8  = bperm_f((int)lane + 16, d2b[0]);  // thetaR row 8
        const float vt2  = bperm_f((int)lane + 16, vtA);     // vertex feature 2
        const float lt2  = bperm_f((int)lane + 16, ltA);     // label feature 2
        const float vt0 = vtA, vt1 = vtB, lt0 = ltA, lt1 = ltB;

        // dotR[r] = sum_c thetaR[c][r]*vt[c]   (thetaR[c][r] = row c*3+r)
        const float dr0 = d2b[0] * vt0 + d2b[3] * vt1 + d2b[6] * vt2;
        const float dr1 = d2b[1] * vt0 + d2b[4] * vt1 + d2b[7] * vt2;
        const float dr2 = d2b[2] * vt0 + d2b[5] * vt1 + tR8    * vt2;
        const float t0 = kap0 * dr0, t1 = kap1 * dr1, t2 = kap2 * dr2;
        // out[o] = sum_r thetaL[o][r]*tmp[r]   (thetaL[o][r] = row o*3+r)
        const float o0 = d2a[0] * t0 + d2a[1] * t1 + d2a[2] * t2;
        const float o1 = d2a[3] * t0 + d2a[4] * t1 + d2a[5] * t2;
        const float o2 = d2a[6] * t0 + d2a[7] * t1 + tL8    * t2;
        // gamma/DELTA folded with mean(1/8) and final (1/2) -> *0.0625
        const float g = __expf(-(lt0 * lt0 + lt1 * lt1 + lt2 * lt2) * 0.1f) * 0.0625f;

        if (lo) {
            lds_go[wave][3u * s + 0u] = g * o0;
            lds_go[wave][3u * s + 1u] = g * o1;
            lds_go[wave][3u * s + 2u] = g * o2;
        }
        __syncthreads();
        // 48 flat values -> exactly 6 exclusively-owned output cells (mean over K=8)
        if (lane < 6u) {
            float sum = 0.0f;
#pragma unroll
            for (unsigned i = 0; i < 8u; ++i) sum += lds_go[wave][lane * 8u + i];
            out[tile * 6u + lane] += sum;  // conv kernel already wrote base value
        }
        __syncthreads();                   // lds_go reuse across iterations
    }
}

extern "C" void kernel_launch(void* const* d_in, const int* in_sizes, int n_in,
                              void* d_out, int out_size, void* d_ws, size_t ws_size,
                              hipStream_t stream) {
    const float* h    = (const float*)d_in[0];
    const int*   edge = (const int*)d_in[1];
    const float* W0p  = (const float*)d_in[2];
    const float* b0p  = (const float*)d_in[3];
    const float* WLp  = (const float*)d_in[4];
    const float* bLp  = (const float*)d_in[5];
    const float* WRp  = (const float*)d_in[6];
    const float* bRp  = (const float*)d_in[7];
    const float* Wkp  = (const float*)d_in[8];
    const float* bkp  = (const float*)d_in[9];
    const float* cw   = (const float*)d_in[10];
    const float* bias = (const float*)d_in[11];
    float* out = (float*)d_out;

    // 1) local branch writes base value (initializes every output cell)
    const int convBlocks = out_size / 256;            // (B*COUT*H) = 6144
    conv_local_kernel<<<convBlocks, 256, 0, stream>>>(h, cw, bias, out);

    // 2) non-local branch accumulates on top (exclusive cells, no atomics)
    const int rows = in_sizes[1];                     // B0 = B*N*K = 4,194,304
    const int graphBlocks = rows / (128 * TILES);     // 8 waves x 16 rows x TILES
    graph_branch_kernel<<<graphBlocks, 256, 0, stream>>>(
        h, edge, W0p, b0p, WLp, bLp, WRp, bRp, Wkp, bkp, out);
}